// CMRF_86234353369844
// MI455X (gfx1250) — compile-verified
//
#include <hip/hip_runtime.h>
#include <hip/hip_bf16.h>
#include <stdint.h>

// Problem constants (fixed by the reference).
#define S_DIM 512
#define D_DIM 4096
#define F_DIM 16384
#define NTHREADS 256
#define NCHUNK (F_DIM / NTHREADS)   // 64 chunks of 256 factors

// ---------------------------------------------------------------------------
// CDNA5 async global->LDS copy helpers (gfx1250: GLOBAL_LOAD_ASYNC_TO_LDS_*,
// tracked with ASYNCcnt; completion is in-order, so a counted
// s_wait_asynccnt implements double buffering).
// ---------------------------------------------------------------------------
__device__ __forceinline__ uint32_t lds_off(const void* p) {
  // Generic LDS pointer = {SHARED_BASE aperture, offset[31:0]} -> low 32 bits
  // are the wave-relative LDS byte address the instruction wants.
  return (uint32_t)(uintptr_t)p;
}
__device__ __forceinline__ void async_ld_b128(uint32_t lds, const void* g) {
  asm volatile("global_load_async_to_lds_b128 %0, %1, off"
               :: "v"(lds), "v"((unsigned long long)(uintptr_t)g) : "memory");
}
__device__ __forceinline__ void async_ld_b64(uint32_t lds, const void* g) {
  asm volatile("global_load_async_to_lds_b64 %0, %1, off"
               :: "v"(lds), "v"((unsigned long long)(uintptr_t)g) : "memory");
}
#define WAIT_ASYNC_LE_4() asm volatile("s_wait_asynccnt 4" ::: "memory")
#define WAIT_ASYNC_LE_0() asm volatile("s_wait_asynccnt 0" ::: "memory")

// Fast reciprocal (v_rcp_f32, ~1 ulp) — avoids the IEEE div-scale/fmas chain.
__device__ __forceinline__ float frcp(float v) { return __builtin_amdgcn_rcpf(v); }

// ---------------------------------------------------------------------------
// Kernel 1: one workgroup per sample s.
//   LDS: sigmoid row (16KB), Dx accumulator row (16KB),
//        double-buffered A (16KB) / B (8KB) / vlabel (4KB) staging.  ~60KB
//   -> ~5 workgroups (40 wave32) per WGP: plenty of latency hiding for the
//      v_exp_f32 pipeline and the LDS gather/scatter traffic.
// ---------------------------------------------------------------------------
__global__ __launch_bounds__(NTHREADS)
void cmrf_main_kernel(const float* __restrict__ x,
                      const float* __restrict__ lam_p,
                      const float* __restrict__ A,    // (F,4,2) contiguous
                      const float* __restrict__ Bm,   // (F,4)
                      const int*   __restrict__ G,    // vlabel (F,2)
                      float* __restrict__ w_raw,      // (S,) unnormalized
                      float* __restrict__ Dx)         // (S,D)
{
  __shared__ __align__(16) float lds_sig[D_DIM];
  __shared__ __align__(16) float lds_dx [D_DIM];
  __shared__ __align__(16) float lds_A[2][NTHREADS * 8];
  __shared__ __align__(16) float lds_B[2][NTHREADS * 4];
  __shared__ __align__(16) int   lds_G[2][NTHREADS * 2];
  __shared__ float red_ph[8], red_pd[8];

  const int s   = blockIdx.x;
  const int tid = threadIdx.x;
  const float lam = lam_p[0];
  // disc(x) = (x > 0.5) recovered from sigmoid monotonicity: sig > sigmoid(lam/2)
  const float thr = frcp(1.0f + __expf(-0.5f * lam));

  // ---- Phase 1: sigmoid row + zero the Dx accumulator (vectorized b128) ----
  {
    const float4* xrow4 = (const float4*)(x + (size_t)s * D_DIM);
    float4* sig4 = (float4*)lds_sig;
    float4* dx4  = (float4*)lds_dx;
    #pragma unroll
    for (int i = tid; i < D_DIM / 4; i += NTHREADS) {
      float4 xv = xrow4[i];
      float4 sg;
      sg.x = frcp(1.0f + __expf(-lam * xv.x));
      sg.y = frcp(1.0f + __expf(-lam * xv.y));
      sg.z = frcp(1.0f + __expf(-lam * xv.z));
      sg.w = frcp(1.0f + __expf(-lam * xv.w));
      sig4[i] = sg;
      dx4[i]  = make_float4(0.f, 0.f, 0.f, 0.f);
    }
  }

  // ---- Phase 2: stream factors with async double buffering ----
  auto issue = [&](int c, int b) {
    const char* gA = (const char*)A  + (size_t)c * (NTHREADS * 32); // 8 f32/f
    const char* gB = (const char*)Bm + (size_t)c * (NTHREADS * 16); // 4 f32/f
    const char* gG = (const char*)G  + (size_t)c * (NTHREADS * 8);  // 2 i32/f
    char* lA = (char*)&lds_A[b][0];
    char* lB = (char*)&lds_B[b][0];
    char* lG = (char*)&lds_G[b][0];
    async_ld_b128(lds_off(lA + tid * 16),        gA + tid * 16);
    async_ld_b128(lds_off(lA + 4096 + tid * 16), gA + 4096 + tid * 16);
    async_ld_b128(lds_off(lB + tid * 16),        gB + tid * 16);
    async_ld_b64 (lds_off(lG + tid * 8),         gG + tid * 8);
  };

  float prodPh = 1.0f;  // per-lane partial product of Ph over this lane's f's
  float prodPd = 1.0f;  // same for the binarized path (P)

  issue(0, 0);
  for (int c = 0; c < NCHUNK; ++c) {
    const int b = c & 1;
    if (c + 1 < NCHUNK) { issue(c + 1, (c + 1) & 1); WAIT_ASYNC_LE_4(); }
    else                { WAIT_ASYNC_LE_0(); }
    __syncthreads();   // chunk c staged by all waves; phase-1 LDS visible too

    // Lane handles factor f = c*256 + tid.
    const float4 a0 = ((const float4*)lds_A[b])[tid * 2 + 0]; // A[f,0,:],A[f,1,:]
    const float4 a1 = ((const float4*)lds_A[b])[tid * 2 + 1]; // A[f,2,:],A[f,3,:]
    const float4 bf = ((const float4*)lds_B[b])[tid];         // B[f,0..3]
    const int2   gg = ((const int2 *)lds_G[b])[tid];          // vlabel[f,0..1]

    const float sg0 = lds_sig[gg.x];
    const float sg1 = lds_sig[gg.y];
    const float dg0 = (sg0 > thr) ? 1.0f : 0.0f;
    const float dg1 = (sg1 > thr) ? 1.0f : 0.0f;

    // expterm[f,r,s] = exp(A[f,r,0]*sg0 + A[f,r,1]*sg1 + B[f,r])
    const float e0 = __expf(fmaf(a0.x, sg0, fmaf(a0.y, sg1, bf.x)));
    const float e1 = __expf(fmaf(a0.z, sg0, fmaf(a0.w, sg1, bf.y)));
    const float e2 = __expf(fmaf(a1.x, sg0, fmaf(a1.y, sg1, bf.z)));
    const float e3 = __expf(fmaf(a1.z, sg0, fmaf(a1.w, sg1, bf.w)));
    const float ph = (e0 + e1) + (e2 + e3);
    prodPh *= ph;

    const float q0 = __expf(fmaf(a0.x, dg0, fmaf(a0.y, dg1, bf.x)));
    const float q1 = __expf(fmaf(a0.z, dg0, fmaf(a0.w, dg1, bf.y)));
    const float q2 = __expf(fmaf(a1.x, dg0, fmaf(a1.y, dg1, bf.z)));
    const float q3 = __expf(fmaf(a1.z, dg0, fmaf(a1.w, dg1, bf.w)));
    prodPd *= (q0 + q1) + (q2 + q3);

    // contrib[s,f,v] = (sum_r A[f,r,v]*e_r) / Ph ; scatter into Dx row.
    const float rph = frcp(ph);
    const float c0 = fmaf(a0.x, e0, fmaf(a0.z, e1, fmaf(a1.x, e2, a1.z * e3))) * rph;
    const float c1 = fmaf(a0.y, e0, fmaf(a0.w, e1, fmaf(a1.y, e2, a1.w * e3))) * rph;
    atomicAdd(&lds_dx[gg.x], c0);   // ds_add_f32 (indices may collide)
    atomicAdd(&lds_dx[gg.y], c1);
    __syncthreads();   // chunk c fully consumed -> its buffer reusable at c+2
  }

  // ---- Phase 3: reduce products over 256 lanes (wave32 shuffles + LDS) ----
  #pragma unroll
  for (int off = 16; off > 0; off >>= 1) {
    prodPh *= __shfl_xor(prodPh, off, 32);
    prodPd *= __shfl_xor(prodPd, off, 32);
  }
  const int wid = tid >> 5, lane = tid & 31;
  if (lane == 0) { red_ph[wid] = prodPh; red_pd[wid] = prodPd; }
  __syncthreads();
  if (tid == 0) {
    float ph = 1.0f, pd = 1.0f;
    #pragma unroll
    for (int w = 0; w < 8; ++w) { ph *= red_ph[w]; pd *= red_pd[w]; }
    w_raw[s] = ph * frcp(pd);         // Phat/P, normalized in kernel 2
  }

  // ---- Phase 4: Dx[s,d] = acc[d] * lam*sig*(1-sig), coalesced b128 ----
  {
    const float4* sig4 = (const float4*)lds_sig;
    const float4* dx4  = (const float4*)lds_dx;
    float4* out4 = (float4*)(Dx + (size_t)s * D_DIM);
    #pragma unroll
    for (int i = tid; i < D_DIM / 4; i += NTHREADS) {
      const float4 sg = sig4[i];
      const float4 ac = dx4[i];
      float4 r;
      r.x = ac.x * (lam * sg.x * (1.0f - sg.x));
      r.y = ac.y * (lam * sg.y * (1.0f - sg.y));
      r.z = ac.z * (lam * sg.z * (1.0f - sg.z));
      r.w = ac.w * (lam * sg.w * (1.0f - sg.w));
      out4[i] = r;
    }
  }
}

// ---------------------------------------------------------------------------
// Kernel 2: normalize w_s = w_raw / sum(w_raw). One block, 512 threads.
// ---------------------------------------------------------------------------
__global__ __launch_bounds__(S_DIM)
void cmrf_norm_kernel(const float* __restrict__ w_raw, float* __restrict__ w_out) {
  __shared__ float red[16];
  const int tid = threadIdx.x;
  const float v = w_raw[tid];
  float acc = v;
  #pragma unroll
  for (int off = 16; off > 0; off >>= 1) acc += __shfl_xor(acc, off, 32);
  if ((tid & 31) == 0) red[tid >> 5] = acc;
  __syncthreads();
  if (tid < 16) {
    float r = red[tid];
    #pragma unroll
    for (int off = 8; off > 0; off >>= 1) r += __shfl_xor(r, off, 32);
    if (tid == 0) red[0] = frcp(r);
  }
  __syncthreads();
  w_out[tid] = v * red[0];
}

// ---------------------------------------------------------------------------
extern "C" void kernel_launch(void* const* d_in, const int* in_sizes, int n_in,
                              void* d_out, int out_size, void* d_ws, size_t ws_size,
                              hipStream_t stream) {
  const float* x      = (const float*)d_in[0];   // (S,D)
  const float* lam    = (const float*)d_in[1];   // scalar
  const float* A      = (const float*)d_in[2];   // (F,4,2)
  const float* Bm     = (const float*)d_in[3];   // (F,4)
  const int*   vlabel = (const int*)d_in[4];     // (F,2)

  float* out   = (float*)d_out;
  float* w_out = out;                 // first 512 floats: w_s
  float* Dx    = out + S_DIM;         // then (S,D) Dx
  float* w_raw = (float*)d_ws;        // 512-float scratch

  cmrf_main_kernel<<<dim3(S_DIM), dim3(NTHREADS), 0, stream>>>(
      x, lam, A, Bm, vlabel, w_raw, Dx);
  cmrf_norm_kernel<<<dim3(1), dim3(S_DIM), 0, stream>>>(w_raw, w_out);
}